// CrossDimensionalRefmntNet_48284022341778
// MI455X (gfx1250) — compile-verified
//
#include <hip/hip_runtime.h>
#include <cstddef>

// ---------------- static problem config (matches reference) ----------------
#define N_IMGS    9
#define C_FEAT    24
#define HF        112
#define WF        112
#define HD        56
#define WD        56
#define N_PLANES  64
#define EPR       8          // edges per ref
#define N_EDGES   72
#define N_PIX     (HD*WD)            // 3136
#define CH_STRIDE (HF*WF)            // 12544
#define IMG_STRIDE (C_FEAT*CH_STRIDE)
#define OUT_CSTRIDE (N_PLANES*N_PIX) // 200704
#define OUT_RSTRIDE (C_FEAT*OUT_CSTRIDE)
#define PIX_STEP  (447.0f/55.0f)     // linspace(0,447,56) step
#define SAMP_SCALE (111.0f/447.0f)   // maps proj/z -> feature-map pixel coords
#define GROUPS    (N_PIX/16)         // 196 point-groups of 16
#define WAVE_TASKS (N_IMGS*N_PLANES*GROUPS)   // 112896
#define BLOCKS    (WAVE_TASKS/8)              // 14112

typedef __attribute__((ext_vector_type(2))) float v2f;
typedef __attribute__((ext_vector_type(8))) float v8f;

// ---------------- setup: per-edge affine projection  proj = depth*H*[u,v,1] + b ----
__global__ void cdr_setup(const float* __restrict__ rot,   // [9,3,3]
                          const float* __restrict__ tvec,  // [9,3]
                          const float* __restrict__ Kin,   // [9,3,3]
                          const int*   __restrict__ edges, // [2,72]
                          float* __restrict__ wsH,         // [72,9]
                          float* __restrict__ wsBv,        // [72,3]
                          int*   __restrict__ wsSrc)       // [72]
{
    const int e = blockIdx.x * blockDim.x + threadIdx.x;
    if (e >= N_EDGES) return;
    const int rf = edges[e];
    const int sf = edges[N_EDGES + e];

    float Ks[9], Kr[9], Rs[9], Rr[9];
#pragma unroll
    for (int i = 0; i < 9; ++i) {
        Ks[i] = Kin[sf*9+i]; Kr[i] = Kin[rf*9+i];
        Rs[i] = rot[sf*9+i]; Rr[i] = rot[rf*9+i];
    }
    float ts[3], tr[3];
#pragma unroll
    for (int i = 0; i < 3; ++i) { ts[i] = tvec[sf*3+i]; tr[i] = tvec[rf*3+i]; }

    // Kr^{-1} via adjugate
    float Ki[9];
    {
        const float a=Kr[0],b=Kr[1],c=Kr[2],d=Kr[3],ee=Kr[4],f=Kr[5],g=Kr[6],h=Kr[7],i9=Kr[8];
        const float det = a*(ee*i9-f*h) - b*(d*i9-f*g) + c*(d*h-ee*g);
        const float id = 1.0f/det;
        Ki[0]=(ee*i9-f*h)*id; Ki[1]=(c*h-b*i9)*id; Ki[2]=(b*f-c*ee)*id;
        Ki[3]=(f*g-d*i9)*id;  Ki[4]=(a*i9-c*g)*id; Ki[5]=(c*d-a*f)*id;
        Ki[6]=(d*h-ee*g)*id;  Ki[7]=(b*g-a*h)*id;  Ki[8]=(a*ee-b*d)*id;
    }
    // M = Rs * Rr^T
    float M[9];
#pragma unroll
    for (int i = 0; i < 3; ++i)
#pragma unroll
        for (int j = 0; j < 3; ++j)
            M[i*3+j] = Rs[i*3+0]*Rr[j*3+0] + Rs[i*3+1]*Rr[j*3+1] + Rs[i*3+2]*Rr[j*3+2];
    // KM = Ks * M ;  H = KM * Ki
    float KM[9], H[9];
#pragma unroll
    for (int i = 0; i < 3; ++i)
#pragma unroll
        for (int j = 0; j < 3; ++j)
            KM[i*3+j] = Ks[i*3+0]*M[0*3+j] + Ks[i*3+1]*M[1*3+j] + Ks[i*3+2]*M[2*3+j];
#pragma unroll
    for (int i = 0; i < 3; ++i)
#pragma unroll
        for (int j = 0; j < 3; ++j)
            H[i*3+j] = KM[i*3+0]*Ki[0*3+j] + KM[i*3+1]*Ki[1*3+j] + KM[i*3+2]*Ki[2*3+j];
    // b = Ks * (ts - M*tr)
    float mt[3], dt[3];
#pragma unroll
    for (int i = 0; i < 3; ++i)
        mt[i] = M[i*3+0]*tr[0] + M[i*3+1]*tr[1] + M[i*3+2]*tr[2];
#pragma unroll
    for (int i = 0; i < 3; ++i) dt[i] = ts[i] - mt[i];
#pragma unroll
    for (int i = 0; i < 3; ++i)
        wsBv[e*3+i] = Ks[i*3+0]*dt[0] + Ks[i*3+1]*dt[1] + Ks[i*3+2]*dt[2];
#pragma unroll
    for (int i = 0; i < 9; ++i) wsH[e*9+i] = H[i];
    wsSrc[e] = sf;
}

// ---------------- bilinear sample of one edge's projection, accumulate 24 channels ----
__device__ __forceinline__ void sample_edge(const float* __restrict__ fb,
                                            float PX, float PY, float PZ,
                                            float acc[C_FEAT], float accsq[C_FEAT])
{
    const float z   = fabsf(PZ) + 1e-8f;
    // 1-ulp hardware reciprocal (v_rcp_f32) instead of the IEEE divide expansion:
    // only feeds bilinear sample coordinates, precision is ample.
    const float rz  = __builtin_amdgcn_rcpf(z) * SAMP_SCALE;
    const float x   = PX * rz;
    const float y   = PY * rz;
    const float x0f = floorf(x), y0f = floorf(y);
    const float x1f = x0f + 1.0f, y1f = y0f + 1.0f;
    const float wx  = x - x0f,    wy  = y - y0f;
    const bool vx0 = (x0f >= 0.0f) && (x0f <= (float)(WF-1));
    const bool vx1 = (x1f >= 0.0f) && (x1f <= (float)(WF-1));
    const bool vy0 = (y0f >= 0.0f) && (y0f <= (float)(HF-1));
    const bool vy1 = (y1f >= 0.0f) && (y1f <= (float)(HF-1));
    const float w00 = (vx0 && vy0) ? (1.0f-wx)*(1.0f-wy) : 0.0f;
    const float w10 = (vx1 && vy0) ? wx*(1.0f-wy)        : 0.0f;
    const float w01 = (vx0 && vy1) ? (1.0f-wx)*wy        : 0.0f;
    const float w11 = (vx1 && vy1) ? wx*wy               : 0.0f;
    const int xi0 = (int)fminf(fmaxf(x0f, 0.0f), (float)(WF-1));
    const int xi1 = (int)fminf(fmaxf(x1f, 0.0f), (float)(WF-1));
    const int yi0 = (int)fminf(fmaxf(y0f, 0.0f), (float)(HF-1));
    const int yi1 = (int)fminf(fmaxf(y1f, 0.0f), (float)(HF-1));
    const int i00 = yi0*WF + xi0, i10 = yi0*WF + xi1;
    const int i01 = yi1*WF + xi0, i11 = yi1*WF + xi1;
#pragma unroll
    for (int c = 0; c < C_FEAT; ++c) {
        const float* fc = fb + c*CH_STRIDE;
        const float v = w00*fc[i00] + w10*fc[i10] + w01*fc[i01] + w11*fc[i11];
        acc[c]  += v;
        accsq[c] = fmaf(v, v, accsq[c]);
    }
}

// Build one A operand (16x4 f32 A-matrix layout: lanes0-15 hold K0,K1; lanes16-31 hold K2,K3).
// Rows: {0-2,3-5}=edges base+0,base+1 ; {8-10,11-13}=edges base+2,base+3 ; rest zero.
__device__ __forceinline__ v2f build_a(const float* __restrict__ wsH,
                                       const float* __restrict__ wsBv,
                                       int geBase, int le, int comp, int hi, float depth)
{
    v2f a; a.x = 0.0f; a.y = 0.0f;
    if (le >= 0) {
        const int ge = geBase + le;
        const float* Hrow = wsH + ge*9 + comp*3;
        if (hi == 0) { a.x = depth*Hrow[0]; a.y = depth*Hrow[1]; }
        else         { a.x = depth*Hrow[2]; a.y = wsBv[ge*3 + comp]; }
    }
    return a;
}

// ---------------- fused plane-sweep variance kernel ----------------
__global__ __launch_bounds__(256)
void cdr_fused(const float* __restrict__ feats,   // [9,24,112,112]
               const float* __restrict__ wsH,     // [72,9]
               const float* __restrict__ wsBv,    // [72,3]
               const int*   __restrict__ wsSrc,   // [72]
               float* __restrict__ out)           // [9,24,64,56,56]
{
    const int lane = threadIdx.x & 31;
    int w = blockIdx.x * 8 + (threadIdx.x >> 5);   // wave-task id
    const int g  = w % GROUPS;  w /= GROUPS;
    const int dp = w % N_PLANES;
    const int r  = w / N_PLANES;
    const float depth = 0.5f + 0.05f * (float)dp;

    const int m  = lane & 15;   // WMMA column (point) / A-row index
    const int hi = lane >> 4;   // which K-half / which D-row-half this lane holds

    // This lane's point (column m of the WMMA)
    const int p  = g*16 + m;
    const int pi = p / WD, pj = p % WD;
    const float u = (float)pj * PIX_STEP;
    const float v = (float)pi * PIX_STEP;

    // B operand: columns = [u, v, 1, 1]^T per point. lanes0-15: (K0,K1)=(u,v); lanes16-31: (K2,K3)=(1,1)
    v2f bOp;
    bOp.x = hi ? 1.0f : u;
    bOp.y = hi ? 1.0f : v;

    // A row -> (local edge, proj component)
    int le = -1, comp = 0;
    if (m < 6)                  { le = m / 3;           comp = m - le*3; }
    else if (m >= 8 && m < 14)  { const int mm = m - 8; le = 2 + mm/3;  comp = mm - (mm/3)*3; }

    const v2f aOp0 = build_a(wsH, wsBv, r*EPR,     le, comp, hi, depth);
    const v2f aOp1 = build_a(wsH, wsBv, r*EPR + 4, le, comp, hi, depth);

    float acc[C_FEAT], accsq[C_FEAT];
#pragma unroll
    for (int c = 0; c < C_FEAT; ++c) { acc[c] = 0.0f; accsq[c] = 0.0f; }

    const int e0 = r*EPR + 2*hi;  // this lane's first edge (D rows it receives)
    const v8f cz = {};

    // WMMA #1: edges r*8 .. r*8+3  (lane gets edges e0, e0+1 in D[0..5])
    v8f dres = __builtin_amdgcn_wmma_f32_16x16x4_f32(
        false, aOp0, false, bOp, (short)0, cz, false, false);
    {
        const float* f0 = feats + (size_t)wsSrc[e0    ] * IMG_STRIDE;
        const float* f1 = feats + (size_t)wsSrc[e0 + 1] * IMG_STRIDE;
        sample_edge(f0, dres[0], dres[1], dres[2], acc, accsq);
        sample_edge(f1, dres[3], dres[4], dres[5], acc, accsq);
    }
    // WMMA #2: edges r*8+4 .. r*8+7 (lane gets edges e0+4, e0+5)
    dres = __builtin_amdgcn_wmma_f32_16x16x4_f32(
        false, aOp1, false, bOp, (short)0, cz, false, false);
    {
        const float* f2 = feats + (size_t)wsSrc[e0 + 4] * IMG_STRIDE;
        const float* f3 = feats + (size_t)wsSrc[e0 + 5] * IMG_STRIDE;
        sample_edge(f2, dres[0], dres[1], dres[2], acc, accsq);
        sample_edge(f3, dres[3], dres[4], dres[5], acc, accsq);
    }

    // Combine lane pair (lanes n and n+16 hold partial sums over disjoint edge sets of the same point)
#pragma unroll
    for (int c = 0; c < C_FEAT; ++c) {
        acc[c]   += __shfl_xor(acc[c],   16, 32);
        accsq[c] += __shfl_xor(accsq[c], 16, 32);
    }

    // Variance write (denominator is exactly 8 edges per ref); lower half writes all 24 channels
    if (hi == 0) {
        const size_t ob = (size_t)r*OUT_RSTRIDE + (size_t)dp*N_PIX + (size_t)p;
#pragma unroll
        for (int c = 0; c < C_FEAT; ++c) {
            const float mean = acc[c] * 0.125f;
            out[ob + (size_t)c*OUT_CSTRIDE] = fmaf(-mean, mean, accsq[c]*0.125f);
        }
    }
}

// ---------------- launch ----------------
extern "C" void kernel_launch(void* const* d_in, const int* in_sizes, int n_in,
                              void* d_out, int out_size, void* d_ws, size_t ws_size,
                              hipStream_t stream)
{
    const float* feats = (const float*)d_in[0];
    const float* rot   = (const float*)d_in[1];
    const float* tvec  = (const float*)d_in[2];
    const float* Kmat  = (const float*)d_in[3];
    const int*   edges = (const int*)d_in[4];

    float* wsH  = (float*)d_ws;            // 72*9 floats
    float* wsBv = wsH + N_EDGES*9;         // 72*3 floats
    int*   wsSrc = (int*)(wsBv + N_EDGES*3);

    cdr_setup<<<1, 128, 0, stream>>>(rot, tvec, Kmat, edges, wsH, wsBv, wsSrc);
    cdr_fused<<<BLOCKS, 256, 0, stream>>>(feats, wsH, wsBv, wsSrc, (float*)d_out);
}